// QuantumLayer_66211215835284
// MI455X (gfx1250) — compile-verified
//
#include <hip/hip_runtime.h>
#include <hip/hip_bf16.h>
#include <math.h>

typedef float v2f __attribute__((ext_vector_type(2)));
typedef float v8f __attribute__((ext_vector_type(8)));

#define NQ  4
#define DIM 16
#define WAVES_PER_BLOCK 8
#define TILES_PER_WAVE  8

// ---------------------------------------------------------------------------
// Setup: build M = U * diag((-i)^popcount) on device from the (3,4) weights.
// One wave; lanes 0..15 each simulate one basis column through the fixed
// part of the circuit (3 layers of RX(w) + CNOT-ring permutations).
// ws layout: ReM[b*16+k] at ws[0..255], ImM[b*16+k] at ws[256..511].
// ---------------------------------------------------------------------------
__global__ void qsetup_kernel(const float* __restrict__ w, float* __restrict__ ws) {
  int k = threadIdx.x;
  if (k >= DIM) return;
  float re[DIM], im[DIM];
#pragma unroll
  for (int i = 0; i < DIM; ++i) { re[i] = (i == k) ? 1.f : 0.f; im[i] = 0.f; }
  for (int l = 0; l < 3; ++l) {
#pragma unroll
    for (int q = 0; q < NQ; ++q) {
      float sn, cs;
      __sincosf(0.5f * w[l * NQ + q], &sn, &cs);
      const int bit = 8 >> q;            // qubit q <-> index bit (3-q)
#pragma unroll
      for (int i = 0; i < DIM; ++i) {
        if (i & bit) continue;
        const int j = i | bit;
        float r0 = re[i], i0 = im[i], r1 = re[j], i1 = im[j];
        // RX: [[c, -i s],[-i s, c]]
        re[i] = cs * r0 + sn * i1;  im[i] = cs * i0 - sn * r1;
        re[j] = cs * r1 + sn * i0;  im[j] = cs * i1 - sn * r0;
      }
    }
#pragma unroll
    for (int q = 0; q < NQ; ++q) {       // CNOT(q, (q+1)%4) permutation
      const int t = (q + 1) & 3;
      float tr[DIM], ti[DIM];
#pragma unroll
      for (int i = 0; i < DIM; ++i) {
        const int ctrl = (i >> (3 - q)) & 1;
        const int src  = i ^ (ctrl << (3 - t));
        tr[i] = re[src]; ti[i] = im[src];
      }
#pragma unroll
      for (int i = 0; i < DIM; ++i) { re[i] = tr[i]; im[i] = ti[i]; }
    }
  }
  // column phase (-i)^{popcount(k)}
  const int p = __popc(k) & 3;
  const float pr = (p == 0) ? 1.f : (p == 2) ? -1.f : 0.f;
  const float pi = (p == 1) ? -1.f : (p == 3) ? 1.f : 0.f;
  float* ReM = ws;
  float* ImM = ws + 256;
#pragma unroll
  for (int b = 0; b < DIM; ++b) {
    ReM[b * DIM + k] = pr * re[b] - pi * im[b];
    ImM[b * DIM + k] = pr * im[b] + pi * re[b];
  }
}

// ---------------------------------------------------------------------------
// Main: each wave loops over TILES_PER_WAVE tiles of 16 samples.
// Loop-invariant (hoisted, lives in VGPRs): B-fragments of ReM^T, ImM^T and
// the +/-1 sign matrix. Per tile:
//  Fr/Fi = R x ReM^T / ImM^T   (8x v_wmma_f32_16x16x4_f32, K=16 in 4 chunks)
//  P     = Fr^2 + Fi^2          (elementwise, D-layout)
//  out   = P x S^T              (4x WMMA after a wave-private LDS transpose)
// ---------------------------------------------------------------------------
__global__ void __launch_bounds__(256) qmain_kernel(
    const float* __restrict__ x, const float* __restrict__ ws,
    float* __restrict__ out, int nsamp, int ntiles) {
  __shared__ float lds[WAVES_PER_BLOCK][DIM * DIM];   // 1KB per wave
  const int lane = threadIdx.x & 31;
  const int wid  = threadIdx.x >> 5;
  const int half = lane >> 4;                // 0: K rows 0..1, 1: rows 2..3
  const int l15  = lane & 15;
  const int gwave = blockIdx.x * WAVES_PER_BLOCK + wid;
  const int tile0 = gwave * TILES_PER_WAVE;

  const float* ReM = ws;
  const float* ImM = ws + 256;

  // ---- loop-invariant B fragments (4x16 f32: VGPR0 = rows K0,K2; VGPR1 = K1,K3)
  v2f Bre[4], Bim[4], Bs[4];
#pragma unroll
  for (int j = 0; j < 4; ++j) {
    const int k0 = 4 * j + 2 * half;
    Bre[j] = *(const v2f*)(ReM + l15 * DIM + k0);
    Bim[j] = *(const v2f*)(ImM + l15 * DIM + k0);
    float s0 = 0.f, s1 = 0.f;
    if (l15 < NQ) {                  // sign matrix S[q,b] = 1 - 2*bit_{3-q}(b)
      s0 = 1.f - 2.f * (float)((k0 >> (3 - l15)) & 1);
      s1 = 1.f - 2.f * (float)(((k0 + 1) >> (3 - l15)) & 1);
    }
    Bs[j][0] = s0; Bs[j][1] = s1;
  }

  float* lbuf = lds[wid];
  float* st_ptr = lbuf + (8 * half) * DIM + l15;   // element e -> st_ptr[e*16]
  const float* ld_ptr = lbuf + l15 * DIM + 2 * half;

  for (int t = 0; t < TILES_PER_WAVE; ++t) {
    const int tile = tile0 + t;                // wave-uniform
    if (tile >= ntiles) break;                 // uniform exit: EXEC stays full

    // full tile? (wave-uniform; only the very last tile can be partial)
    const bool full = (tile * 16 + 16 <= nsamp);

    // ---- per-sample feature vector R (tensor product of cos/sin halves)
    const int samp  = tile * 16 + l15;
    const int csamp = full ? samp : ((samp < nsamp) ? samp : (nsamp - 1));
    const float4 xv = *(const float4*)(x + (size_t)csamp * 4);
    float sn0, cs0, sn1, cs1, sn2, cs2, sn3, cs3;
    __sincosf(0.5f * xv.x, &sn0, &cs0);
    __sincosf(0.5f * xv.y, &sn1, &cs1);
    __sincosf(0.5f * xv.z, &sn2, &cs2);
    __sincosf(0.5f * xv.w, &sn3, &cs3);
    const float p01[4] = {cs0 * cs1, cs0 * sn1, sn0 * cs1, sn0 * sn1};
    const float p23l   = (half == 0) ? cs2 * cs3 : sn2 * cs3;
    const float p23h   = (half == 0) ? cs2 * sn3 : sn2 * sn3;
    v2f A[4];
#pragma unroll
    for (int j = 0; j < 4; ++j) {    // R[k] = p01[k>>2] * p23[k&3]
      A[j][0] = p01[j] * p23l;
      A[j][1] = p01[j] * p23h;
    }

    // ---- Fr = R * ReM^T, Fi = R * ImM^T
    v8f Fr = {}; v8f Fi = {};
#pragma unroll
    for (int j = 0; j < 4; ++j) {
      Fr = __builtin_amdgcn_wmma_f32_16x16x4_f32(false, A[j], false, Bre[j],
                                                 (short)0, Fr, false, false);
      Fi = __builtin_amdgcn_wmma_f32_16x16x4_f32(false, A[j], false, Bim[j],
                                                 (short)0, Fi, false, false);
    }

    // ---- P = |final|^2, wave-private LDS transpose (D-layout -> A-layout)
    __builtin_amdgcn_wave_barrier();           // order vs prior iter's loads
#pragma unroll
    for (int e = 0; e < 8; ++e)
      st_ptr[e * DIM] = Fr[e] * Fr[e] + Fi[e] * Fi[e];
    __builtin_amdgcn_wave_barrier();           // DS in-order per wave

    v2f Ap[4];
#pragma unroll
    for (int j = 0; j < 4; ++j)
      Ap[j] = *(const v2f*)(ld_ptr + 4 * j);

    // ---- out = P * S^T
    v8f O = {};
#pragma unroll
    for (int j = 0; j < 4; ++j)
      O = __builtin_amdgcn_wmma_f32_16x16x4_f32(false, Ap[j], false, Bs[j],
                                                (short)0, O, false, false);

    // ---- store: lanes with n = l15 < 4 hold out[sample = e+8*half][n]
    if (full) {
      if (l15 < NQ) {
        float* obase = out + (size_t)(tile * 16 + 8 * half) * NQ + l15;
#pragma unroll
        for (int e = 0; e < 8; ++e)
          obase[e * NQ] = O[e];                // 8 stores, immediate offsets
      }
    } else {
      if (l15 < NQ) {
#pragma unroll
        for (int e = 0; e < 8; ++e) {
          const int s = tile * 16 + e + 8 * half;
          if (s < nsamp) out[(size_t)s * NQ + l15] = O[e];
        }
      }
    }
  }
}

extern "C" void kernel_launch(void* const* d_in, const int* in_sizes, int n_in,
                              void* d_out, int out_size, void* d_ws, size_t ws_size,
                              hipStream_t stream) {
  const float* x = (const float*)d_in[0];   // (B, 4) f32
  const float* w = (const float*)d_in[1];   // (3, 4) f32
  float* out = (float*)d_out;               // (B, 4) f32
  float* ws  = (float*)d_ws;                // 512 floats: ReM | ImM

  const int nsamp  = in_sizes[0] / NQ;
  qsetup_kernel<<<1, 32, 0, stream>>>(w, ws);
  const int ntiles = (nsamp + 15) / 16;
  const int waves  = (ntiles + TILES_PER_WAVE - 1) / TILES_PER_WAVE;
  const int nblocks = (waves + WAVES_PER_BLOCK - 1) / WAVES_PER_BLOCK;
  qmain_kernel<<<nblocks, 256, 0, stream>>>(x, ws, out, nsamp, ntiles);
}